// AdditiveAttention_69853348102385
// MI455X (gfx1250) — compile-verified
//
#include <hip/hip_runtime.h>
#include <math.h>

typedef __attribute__((ext_vector_type(2))) float v2f;
typedef __attribute__((ext_vector_type(4))) float v4f;
typedef __attribute__((ext_vector_type(8))) float v8f;

#define NB   4
#define NH   8
#define LQ   256
#define LK   256
#define DK   64
#define NBH  (NB*NH)          // 32
#define ROWS (NBH*LQ)         // 8192

// gfx1250 has V_TANH_F32 (TRANS op, co-executes with VALU). Guarded fallback.
#if __has_builtin(__builtin_amdgcn_tanhf)
#define TANHF(x) __builtin_amdgcn_tanhf(x)
#elif __has_builtin(__builtin_amdgcn_tanh_f32)
#define TANHF(x) __builtin_amdgcn_tanh_f32(x)
#else
#define TANHF(x) tanhf(x)
#endif

#if __has_builtin(__builtin_amdgcn_exp2f)
#define EXP2F(x) __builtin_amdgcn_exp2f(x)
#else
#define EXP2F(x) exp2f(x)
#endif
#define EXPN(x) EXP2F((x) * 1.4426950408889634f)

// ---------------------------------------------------------------------------
// Kernel 1: P[r, e] = sum_d X[r, d] * W[e, d] + bias[e]   (X: ROWS x 64)
// One wave per 16-row tile; WMMA f32 16x16x4, K-loop of 16, 4 N-tiles.
// ---------------------------------------------------------------------------
__global__ __launch_bounds__(32) void proj_wmma_kernel(
    const float* __restrict__ X, const float* __restrict__ W,
    const float* __restrict__ bias, float* __restrict__ P)
{
  const int lane = threadIdx.x & 31;
  const int m    = lane & 15;       // row (A) / col (B,D)
  const int kh   = lane >> 4;       // K-half select
  const int r0   = blockIdx.x * 16;

  const v8f zero = {0.f,0.f,0.f,0.f,0.f,0.f,0.f,0.f};
  v8f c[4] = {zero, zero, zero, zero};

  for (int kk = 0; kk < 16; ++kk) {
    const int colA = kk * 4 + kh * 2;
    v2f a = *(const v2f*)(X + (r0 + m) * DK + colA);
#pragma unroll
    for (int nt = 0; nt < 4; ++nt) {
      v2f b = *(const v2f*)(W + (nt * 16 + m) * DK + colA);
      c[nt] = __builtin_amdgcn_wmma_f32_16x16x4_f32(
          false, a, false, b, (short)0, c[nt], false, false);
    }
  }
#pragma unroll
  for (int nt = 0; nt < 4; ++nt) {
    const float bb = bias[nt * 16 + m];
#pragma unroll
    for (int j = 0; j < 8; ++j) {
      const int mrow = j + 8 * kh;            // C/D layout: VGPR j -> M=j / M=j+8
      P[(r0 + mrow) * DK + nt * 16 + m] = c[nt][j] + bb;
    }
  }
}

// ---------------------------------------------------------------------------
// Kernel 2: fused scores + softmax + attn@v, K-split for occupancy.
// Grid = 512 blocks (one 16-q tile each); block = 256 threads = 8 waves.
// Each wave owns a 32-wide k-slice: phase B computes its slice's scores
// (lane = q:16 x d-half:2, tanh inner loop from LDS-staged kp) with online
// per-slice softmax stats; stats merged across waves via LDS; phase C builds
// normalized attn in place and accumulates a partial 16x64 output tile with
// WMMA f32 16x16x4; partial tiles reduced across waves via LDS (buffer reuse:
// the 64KB kp stage is dead after phase B and is recycled for stats+partials).
// ---------------------------------------------------------------------------
__global__ __launch_bounds__(256) void attn_fused_kernel(
    const float* __restrict__ qp, const float* __restrict__ kp,
    const float* __restrict__ v,  const float* __restrict__ vsw,
    const float* __restrict__ vsb_p,
    float* __restrict__ out, float* __restrict__ attn)
{
  __shared__ float kpS[LK * DK];              // 64 KB, multipurpose

  const int tid  = threadIdx.x;
  const int bh   = blockIdx.x >> 4;           // 16 q-tiles per head
  const int qt   = blockIdx.x & 15;
  const int wave = tid >> 5;
  const int lane = tid & 31;
  const int qq   = lane & 15;
  const int kh   = lane >> 4;
  const int q0   = qt * 16;
  const int kbase = wave * 32;                // this wave's k-slice

  // Stage kp for this head into LDS (float4, fully coalesced).
  {
    const v4f* src = (const v4f*)(kp + bh * LK * DK);
    v4f* dst = (v4f*)kpS;
    for (int i = tid; i < (LK * DK) / 4; i += 256) dst[i] = src[i];
  }
  __syncthreads();

  const float vsb = vsb_p[0];
  const int dbase = kh * 32;                  // this lane's 32-wide d slice

  v4f qpr[8], vswr[8];
  {
    const v4f* qrow = (const v4f*)(qp + (bh * LQ + q0 + qq) * DK + dbase);
    const v4f* vr   = (const v4f*)(vsw + dbase);
#pragma unroll
    for (int i = 0; i < 8; ++i) { qpr[i] = qrow[i]; vswr[i] = vr[i]; }
  }

  float* attnRow = attn + (bh * LQ + q0 + qq) * LK;   // this lane's q row

  // ---- Phase B: scores over this wave's k-slice + online softmax stats ----
  float mrun = -__builtin_inff();
  float srun = 0.f;
  for (int kk = 0; kk < 32; ++kk) {
    const int k = kbase + kk;
    const v4f* krow = (const v4f*)(kpS + k * DK + dbase);  // 2-addr broadcast
    float acc = 0.f;
#pragma unroll
    for (int i = 0; i < 8; ++i) {
      const v4f kv = krow[i];
      const v4f qv = qpr[i];
      const v4f wv = vswr[i];
      acc = fmaf(wv.x, TANHF(qv.x + kv.x), acc);
      acc = fmaf(wv.y, TANHF(qv.y + kv.y), acc);
      acc = fmaf(wv.z, TANHF(qv.z + kv.z), acc);
      acc = fmaf(wv.w, TANHF(qv.w + kv.w), acc);
    }
    acc += __shfl_xor(acc, 16, 32);           // combine the two d-halves
    const float sc = acc + vsb;
    if (kh == 0) attnRow[k] = sc;             // park raw score in attn output
    const float nm = fmaxf(mrun, sc);
    srun = srun * EXPN(mrun - nm) + EXPN(sc - nm);
    mrun = nm;
  }
  __syncthreads();                            // kp stage now dead

  // ---- Merge softmax stats across the 8 waves (reuse kpS[0..255]) ---------
  if (kh == 0) {
    kpS[wave * 16 + qq]       = mrun;         // per-slice max
    kpS[128 + wave * 16 + qq] = srun;         // per-slice rescaled sum
  }
  __syncthreads();

  float M = -__builtin_inff();
#pragma unroll
  for (int w = 0; w < 8; ++w) M = fmaxf(M, kpS[w * 16 + qq]);
  float S = 0.f;
#pragma unroll
  for (int w = 0; w < 8; ++w)
    S += kpS[128 + w * 16 + qq] * EXPN(kpS[w * 16 + qq] - M);
  const float inv = 1.0f / S;

  // ---- Phase C: normalize in place + partial attn@v via WMMA --------------
  const v8f zero = {0.f,0.f,0.f,0.f,0.f,0.f,0.f,0.f};
  v8f c[4] = {zero, zero, zero, zero};
  const float* vhead = v + bh * LK * DK;

  for (int ks = 0; ks < 8; ++ks) {            // 8 K-steps x K=4 = 32 k
    const int row0 = kbase + ks * 4 + kh * 2; // A-frag K split across halves
    v2f sc2 = *(const v2f*)(attnRow + row0);  // same-wave store->load: ordered
    v2f p;
    p.x = EXPN(sc2.x - M) * inv;
    p.y = EXPN(sc2.y - M) * inv;
    *(v2f*)(attnRow + row0) = p;              // final normalized attn output
#pragma unroll
    for (int nt = 0; nt < 4; ++nt) {
      v2f b;
      b.x = vhead[(row0    ) * DK + nt * 16 + qq];
      b.y = vhead[(row0 + 1) * DK + nt * 16 + qq];
      c[nt] = __builtin_amdgcn_wmma_f32_16x16x4_f32(
          false, p, false, b, (short)0, c[nt], false, false);
    }
  }

  // ---- Reduce partial output tiles across waves (reuse kpS) ---------------
  __syncthreads();                            // all waves done reading stats
#pragma unroll
  for (int nt = 0; nt < 4; ++nt) {
#pragma unroll
    for (int j = 0; j < 8; ++j) {
      const int mrow = j + 8 * kh;
      kpS[wave * 1024 + mrow * 64 + nt * 16 + qq] = c[nt][j];
    }
  }
  __syncthreads();

  float* orow = out + (bh * LQ + q0) * DK;
  for (int idx = tid; idx < 16 * DK; idx += 256) {
    float s = 0.f;
#pragma unroll
    for (int w = 0; w < 8; ++w) s += kpS[w * 1024 + idx];
    orow[(idx >> 6) * DK + (idx & 63)] = s;
  }
}

// ---------------------------------------------------------------------------
extern "C" void kernel_launch(void* const* d_in, const int* in_sizes, int n_in,
                              void* d_out, int out_size, void* d_ws, size_t ws_size,
                              hipStream_t stream) {
  (void)in_sizes; (void)n_in; (void)out_size; (void)ws_size;
  const float* q    = (const float*)d_in[0];
  const float* k    = (const float*)d_in[1];
  const float* v    = (const float*)d_in[2];
  const float* Wq_w = (const float*)d_in[3];
  const float* Wq_b = (const float*)d_in[4];
  const float* Wk_w = (const float*)d_in[5];
  const float* Wk_b = (const float*)d_in[6];
  const float* vs_w = (const float*)d_in[7];
  const float* vs_b = (const float*)d_in[8];

  float* out  = (float*)d_out;                 // [B,H,LQ,DK]
  float* attn = out + (size_t)ROWS * DK;       // [B,H,LQ,LK]

  float* qpWS = (float*)d_ws;                  // 2 MB
  float* kpWS = qpWS + (size_t)ROWS * DK;      // 2 MB

  proj_wmma_kernel<<<dim3(ROWS / 16), 32, 0, stream>>>(q, Wq_w, Wq_b, qpWS);
  proj_wmma_kernel<<<dim3(ROWS / 16), 32, 0, stream>>>(k, Wk_w, Wk_b, kpWS);
  attn_fused_kernel<<<dim3(ROWS / 16), 256, 0, stream>>>(qpWS, kpWS, v,
                                                         vs_w, vs_b, out, attn);
}